// GCNModel_34239479284114
// MI455X (gfx1250) — compile-verified
//
#include <hip/hip_runtime.h>

typedef __attribute__((ext_vector_type(2))) float v2f;
typedef __attribute__((ext_vector_type(8))) float v8f;

// ---------------------------------------------------------------------------
// Degree / normalization kernels
// ---------------------------------------------------------------------------
__global__ void k_init_deg(float* __restrict__ deg, int n) {
    int i = blockIdx.x * blockDim.x + threadIdx.x;
    if (i < n) deg[i] = 1.0f;  // self-loop contributes 1 to in-degree
}

__global__ void k_count_deg(const int* __restrict__ col, float* __restrict__ deg,
                            long long E) {
    long long i = (long long)blockIdx.x * blockDim.x + threadIdx.x;
    long long stride = (long long)gridDim.x * blockDim.x;
    for (; i < E; i += stride) {
        atomicAdd(&deg[col[i]], 1.0f);
    }
}

__global__ void k_dinv(const float* __restrict__ deg, float* __restrict__ dinv, int n) {
    int i = blockIdx.x * blockDim.x + threadIdx.x;
    if (i < n) dinv[i] = rsqrtf(deg[i]);  // deg >= 1 always (self loop)
}

// ---------------------------------------------------------------------------
// WMMA GEMM: H[nrows,COUT] = X[nrows,CIN] @ W[CIN,COUT], fp32 via
// V_WMMA_F32_16X16X4_F32. One wave computes one 16x16 output tile.
// Requires nrows % 16 == 0, CIN % 4 == 0, COUT % 16 == 0 (true here).
// ---------------------------------------------------------------------------
template <int CIN, int COUT>
__global__ void k_wmma_gemm(const float* __restrict__ X, const float* __restrict__ W,
                            float* __restrict__ H) {
    const int lane = threadIdx.x & 31;
    const int wave = threadIdx.x >> 5;
    const int wavesPerBlock = blockDim.x >> 5;
    const int colTile = blockIdx.y * wavesPerBlock + wave;   // wave-uniform
    if (colTile * 16 >= COUT) return;                        // uniform exit
    const int rowTile = blockIdx.x;

    const int row0 = rowTile * 16;
    const int col0 = colTile * 16;
    const int m    = lane & 15;      // row (A) / col (B,C,D) within tile
    const int half = lane >> 4;      // 0: K={0,1}/M=0..7 ; 1: K={2,3}/M=8..15
    const int koff = half * 2;

    const float* __restrict__ xrow = X + (size_t)(row0 + m) * CIN;

    v8f acc = {};
#pragma unroll 4
    for (int k = 0; k < CIN; k += 4) {
        v2f a, b;
        // A (16x4 f32): v0 = K=0|2, v1 = K=1|3 per ISA layout
        a.x = xrow[k + koff];
        a.y = xrow[k + koff + 1];
        // B (4x16 f32): row K striped across lanes (N = m), halves hold K=2,3
        b.x = W[(size_t)(k + koff) * COUT + col0 + m];
        b.y = W[(size_t)(k + koff + 1) * COUT + col0 + m];
        acc = __builtin_amdgcn_wmma_f32_16x16x4_f32(
            /*neg_a=*/false, a, /*neg_b=*/false, b,
            /*c_mod=*/(short)0, acc, /*reuse_a=*/false, /*reuse_b=*/false);
    }

    // C/D layout: VGPR v -> M = v + 8*half, N = m
    float* __restrict__ hp = H + (size_t)col0 + m;
#pragma unroll
    for (int v = 0; v < 8; ++v) {
        int r = row0 + v + half * 8;
        hp[(size_t)r * COUT] = acc[v];
    }
}

// ---------------------------------------------------------------------------
// out[i,c] = dinv[i]^2 * H[i,c] + bias[c]   (self-loop message + bias)
// ---------------------------------------------------------------------------
template <int C>
__global__ void k_init_out(const float* __restrict__ H, const float* __restrict__ dinv,
                           const float* __restrict__ bias, float* __restrict__ out,
                           long long total) {
    long long i = (long long)blockIdx.x * blockDim.x + threadIdx.x;
    if (i < total) {
        int node = (int)(i / C);
        int c    = (int)(i % C);
        float d  = dinv[node];
        out[i] = d * d * H[i] + bias[c];
    }
}

// ---------------------------------------------------------------------------
// Edge scatter: out[col[e],:] += dinv[row[e]]*dinv[col[e]] * H[row[e],:]
// C/4 lanes per edge, float4 gathers, f32 global atomics (RMW in L2).
// ---------------------------------------------------------------------------
template <int C>
__global__ void k_scatter(const float* __restrict__ H, const int* __restrict__ row,
                          const int* __restrict__ col, const float* __restrict__ dinv,
                          float* __restrict__ out, long long E) {
    constexpr int TPE = C / 4;  // threads per edge (float4 granularity)
    long long i = (long long)blockIdx.x * blockDim.x + threadIdx.x;
    long long stride = (long long)gridDim.x * blockDim.x;
    const long long total = E * TPE;
    for (; i < total; i += stride) {
        long long e = i / TPE;
        int c4 = (int)(i % TPE);
        int r = row[e];
        int t = col[e];
        float nrm = dinv[r] * dinv[t];
        float4 hv = ((const float4*)(H + (size_t)r * C))[c4];
        float* o = out + (size_t)t * C + (size_t)c4 * 4;
        atomicAdd(o + 0, nrm * hv.x);
        atomicAdd(o + 1, nrm * hv.y);
        atomicAdd(o + 2, nrm * hv.z);
        atomicAdd(o + 3, nrm * hv.w);
    }
}

__global__ void k_relu(float* __restrict__ x, long long total) {
    long long i = (long long)blockIdx.x * blockDim.x + threadIdx.x;
    if (i < total) x[i] = fmaxf(x[i], 0.0f);
}

// ---------------------------------------------------------------------------
// Host launcher
// ---------------------------------------------------------------------------
extern "C" void kernel_launch(void* const* d_in, const int* in_sizes, int n_in,
                              void* d_out, int out_size, void* d_ws, size_t ws_size,
                              hipStream_t stream) {
    constexpr int CIN = 128, CHID = 128, COUT = 64;

    const float* x  = (const float*)d_in[0];
    const int*   ei = (const int*)d_in[1];
    const float* W1 = (const float*)d_in[2];
    const float* b1 = (const float*)d_in[3];
    const float* W2 = (const float*)d_in[4];
    const float* b2 = (const float*)d_in[5];
    float* out = (float*)d_out;

    const int       n = in_sizes[0] / CIN;        // 100000
    const long long E = in_sizes[1] / 2;          // 1600000
    const int* row = ei;
    const int* col = ei + E;

    // workspace layout: deg[n] | dinv[n] | h[n*128] | agg1[n*128]
    float* deg  = (float*)d_ws;
    float* dinv = deg + n;
    float* h    = dinv + n;                       // reused as h2 (n*64) in layer 2
    float* agg1 = h + (size_t)n * CHID;

    const int B = 256;
    auto blk1 = [&](long long t) { return (unsigned)((t + B - 1) / B); };

    // --- degrees / dinv (shared by both layers) ---
    k_init_deg<<<blk1(n), B, 0, stream>>>(deg, n);
    k_count_deg<<<blk1(E), B, 0, stream>>>(col, deg, E);
    k_dinv<<<blk1(n), B, 0, stream>>>(deg, dinv, n);

    const int rowTiles = n / 16;                  // 6250, exact

    // --- layer 1: h = x @ W1 ---
    k_wmma_gemm<CIN, CHID><<<dim3(rowTiles, 1), 256, 0, stream>>>(x, W1, h);
    const long long tot1 = (long long)n * CHID;
    k_init_out<CHID><<<blk1(tot1), B, 0, stream>>>(h, dinv, b1, agg1, tot1);
    k_scatter<CHID><<<blk1(E * (CHID / 4)), B, 0, stream>>>(h, row, col, dinv, agg1, E);
    k_relu<<<blk1(tot1), B, 0, stream>>>(agg1, tot1);

    // --- layer 2: h2 = relu(agg1) @ W2 ---
    k_wmma_gemm<CHID, COUT><<<dim3(rowTiles, 1), 128, 0, stream>>>(agg1, W2, h);
    const long long tot2 = (long long)n * COUT;
    k_init_out<COUT><<<blk1(tot2), B, 0, stream>>>(h, dinv, b2, out, tot2);
    k_scatter<COUT><<<blk1(E * (COUT / 4)), B, 0, stream>>>(h, row, col, dinv, out, E);
}